// StochasticPool2D_1580547969981
// MI455X (gfx1250) — compile-verified
//
#include <hip/hip_runtime.h>
#include <stdint.h>

// Problem geometry (from reference): x[16][64][128][128] f32, K=3, S=1
#define BB      16
#define CCH     64
#define HH      128
#define WW      128
#define HO      126
#define WO      126
#define TILE_H  14                  // output rows per block
#define IN_ROWS (TILE_H + 2)        // 16 input rows staged in LDS
#define NTILES  (HO / TILE_H)       // 9

// Cheap PCG-style 32-bit hash -> uniform [0,1), keyed by flat output index (+seed 42).
__device__ __forceinline__ float uniform01(uint32_t idx) {
    uint32_t h = idx * 747796405u + 2891336453u + 42u;
    h = ((h >> ((h >> 28) + 4u)) ^ h) * 277803737u;
    h = (h >> 22) ^ h;
    return (float)h * 2.3283064365386963e-10f;   // * 2^-32
}

__global__ __launch_bounds__(256) void stochpool2d_kernel(const float* __restrict__ x,
                                                          float* __restrict__ out) {
    __shared__ float smem[IN_ROWS * WW];          // 16*128*4B = 8 KB

    const int tid   = threadIdx.x;                // 0..255
    const int tile  = blockIdx.x;                 // 0..8   (row strip)
    const int plane = blockIdx.y;                 // 0..1023 (b*C + c)
    const int r0    = tile * TILE_H;              // first input/output row of strip

    // ---------------- async global -> LDS stage (CDNA5 path) ----------------
    // 16 rows x 128 floats = 8192 B = 512 x 16B chunks; 2 chunks/thread.
    const uint64_t gbase = (uint64_t)(uintptr_t)x
                         + ((uint64_t)plane * (HH * WW) + (uint64_t)r0 * WW) * 4ull;
    // Generic address of an LDS object holds the LDS byte offset in bits[31:0].
    const uint32_t lbase = (uint32_t)(uintptr_t)smem;

#pragma unroll
    for (int i = 0; i < 2; ++i) {
        const uint32_t byteoff = (uint32_t)(tid + i * 256) * 16u;
        uint32_t laddr = lbase + byteoff;
        uint64_t gaddr = gbase + byteoff;
        asm volatile("global_load_async_to_lds_b128 %0, %1, off"
                     :
                     : "v"(laddr), "v"(gaddr)
                     : "memory");
    }
    asm volatile("s_wait_asynccnt 0x0" ::: "memory");   // this wave's copies done
    __syncthreads();                                     // LDS visible block-wide

    // ---------------- stencil + stochastic sampling from LDS ----------------
    for (int e = tid; e < TILE_H * WO; e += 256) {
        const int r = e / WO;           // output row within strip
        const int c = e - r * WO;       // output col

        // Load 3x3 window once; relu once; accumulate mass.
        float w[9], rw[9];
        float s = 0.f;
#pragma unroll
        for (int i = 0; i < 3; ++i) {
#pragma unroll
            for (int j = 0; j < 3; ++j) {
                const float v  = smem[(r + i) * WW + (c + j)];
                const float rv = fmaxf(v, 0.f);
                w[i * 3 + j]  = v;
                rw[i * 3 + j] = rv;
                s += rv;
            }
        }

        const uint32_t oid = (uint32_t)((plane * HO + (r0 + r)) * WO + c);
        const float u = uniform01(oid);

        // All-zero window -> uniform over 9: fold in as +1 bias per weight.
        const bool  pos    = (s > 0.f);
        const float bias   = pos ? 0.f : 1.f;
        const float target = u * (pos ? s : 9.0f);

        // Fused inverse-CDF walk + value select:
        // val = w[k], k = first index whose prefix sum exceeds target (clamped to 8).
        float cum = rw[0] + bias;
        float val = w[0];
#pragma unroll
        for (int t = 1; t < 9; ++t) {
            val = (cum <= target) ? w[t] : val;
            if (t < 8) cum += rw[t] + bias;
        }

        __builtin_nontemporal_store(val, &out[oid]);   // write-once stream
    }
}

extern "C" void kernel_launch(void* const* d_in, const int* in_sizes, int n_in,
                              void* d_out, int out_size, void* d_ws, size_t ws_size,
                              hipStream_t stream) {
    const float* x = (const float*)d_in[0];
    float* out = (float*)d_out;
    (void)in_sizes; (void)n_in; (void)out_size; (void)d_ws; (void)ws_size;

    dim3 grid(NTILES, BB * CCH);   // 9 x 1024 blocks
    stochpool2d_kernel<<<grid, 256, 0, stream>>>(x, out);
}